// GNN_83141976917064
// MI455X (gfx1250) — compile-verified
//
#include <hip/hip_runtime.h>

#define NN 50000
#define NE 800000
#define D 128
#define D2 256
#define BN_EPS 1e-5f

typedef float v2f __attribute__((ext_vector_type(2)));
typedef float v8f __attribute__((ext_vector_type(8)));

// h[n,d] = xemb1[x[n,0], d] + xemb2[x[n,1], d]
__global__ void embed_kernel(const int* __restrict__ x,
                             const float* __restrict__ xemb1,
                             const float* __restrict__ xemb2,
                             float* __restrict__ h) {
  int i = blockIdx.x * blockDim.x + threadIdx.x;
  if (i >= NN * D) return;
  int n = i >> 7, d = i & (D - 1);
  h[i] = xemb1[x[2 * n] * D + d] + xemb2[x[2 * n + 1] * D + d];
}

// agg = h + e1l[4,:] + e2l[0,:]   (self-loop contribution; also initializes agg)
__global__ void init_agg_kernel(const float* __restrict__ h,
                                const float* __restrict__ e1l,
                                const float* __restrict__ e2l,
                                float* __restrict__ agg) {
  int i = blockIdx.x * blockDim.x + threadIdx.x;
  if (i >= NN * D) return;
  int d = i & (D - 1);
  agg[i] = h[i] + e1l[4 * D + d] + e2l[d];
}

// one wave per edge: lane handles 4 channels (32*4 = 128)
__global__ void scatter_kernel(const int* __restrict__ ei,
                               const int* __restrict__ ea,
                               const float* __restrict__ h,
                               const float* __restrict__ e1l,
                               const float* __restrict__ e2l,
                               float* __restrict__ agg) {
  int t = blockIdx.x * blockDim.x + threadIdx.x;
  int e = t >> 5;
  if (e >= NE) return;
  int lane = t & 31;
  int src = ei[e];
  int dst = ei[NE + e];
  int ba = ea[2 * e];
  int bb = ea[2 * e + 1];
  int d0 = lane * 4;
  float4 hv = *(const float4*)(h + (size_t)src * D + d0);
  float4 a1 = *(const float4*)(e1l + (size_t)ba * D + d0);
  float4 a2 = *(const float4*)(e2l + (size_t)bb * D + d0);
  float* o = agg + (size_t)dst * D + d0;
  atomicAdd(o + 0, hv.x + a1.x + a2.x);
  atomicAdd(o + 1, hv.y + a1.y + a2.y);
  atomicAdd(o + 2, hv.z + a1.z + a2.z);
  atomicAdd(o + 3, hv.w + a1.w + a2.w);
}

// out[M,Ncol] = act(A[M,K] @ W[K,Ncol] + bias), one wave -> 16 x (16*NT) tile
// V_WMMA_F32_16X16X4_F32 layout:
//   A 16x4 : lanes 0-15 rows M, v[0]=K0 v[1]=K1 ; lanes 16-31 v[0]=K2 v[1]=K3
//   B 4x16 : lanes 0-15 cols N, v[0]=row K0 v[1]=row K1 ; lanes 16-31 K2/K3
//   C/D    : lane = col N (mod 16), vgpr r = row M=r (lanes<16) / M=8+r (lanes>=16)
template <int NT>
__global__ void wmma_gemm_bias(const float* __restrict__ A,
                               const float* __restrict__ W,
                               const float* __restrict__ bias,
                               float* __restrict__ out,
                               int K, int Ncol, int totalWaves, int relu) {
  int wid = (blockIdx.x * blockDim.x + threadIdx.x) >> 5;
  if (wid >= totalWaves) return;
  int lane = threadIdx.x & 31;
  int l16 = lane & 15;
  int half = lane >> 4;              // 0: K+0/1 , 1: K+2/3
  int tilesN = Ncol / (16 * NT);
  int tm = wid / tilesN;
  int tn = wid % tilesN;
  int rowA = tm * 16 + l16;
  int colBase = tn * 16 * NT;

  v8f acc[NT] = {};
  for (int kk = 0; kk < K; kk += 4) {
    v2f a = *(const v2f*)(A + (size_t)rowA * K + kk + half * 2);
#pragma unroll
    for (int t = 0; t < NT; ++t) {
      int col = colBase + t * 16 + l16;
      v2f b;
      b.x = W[(size_t)(kk + half * 2 + 0) * Ncol + col];
      b.y = W[(size_t)(kk + half * 2 + 1) * Ncol + col];
      acc[t] = __builtin_amdgcn_wmma_f32_16x16x4_f32(
          /*neg_a=*/false, a, /*neg_b=*/false, b,
          /*c_mod=*/(short)0, acc[t], /*reuse_a=*/false, /*reuse_b=*/false);
    }
  }
#pragma unroll
  for (int t = 0; t < NT; ++t) {
    int col = colBase + t * 16 + l16;
    float bv = bias[col];
#pragma unroll
    for (int r = 0; r < 8; ++r) {
      int row = tm * 16 + r + half * 8;
      float v = acc[t][r] + bv;
      if (relu) v = fmaxf(v, 0.f);
      out[(size_t)row * Ncol + col] = v;
    }
  }
}

// one block per channel: biased batch stats -> scale/shift
__global__ void bn_stats_kernel(const float* __restrict__ h2,
                                const float* __restrict__ gamma,
                                const float* __restrict__ beta,
                                float* __restrict__ stats) {
  int d = blockIdx.x;
  float s = 0.f, q = 0.f;
  for (int n = threadIdx.x; n < NN; n += blockDim.x) {
    float v = h2[(size_t)n * D + d];
    s += v;
    q += v * v;
  }
  __shared__ float shs[256], shq[256];
  shs[threadIdx.x] = s;
  shq[threadIdx.x] = q;
  __syncthreads();
  for (int off = 128; off > 0; off >>= 1) {
    if (threadIdx.x < off) {
      shs[threadIdx.x] += shs[threadIdx.x + off];
      shq[threadIdx.x] += shq[threadIdx.x + off];
    }
    __syncthreads();
  }
  if (threadIdx.x == 0) {
    float mu = shs[0] / (float)NN;
    float var = shq[0] / (float)NN - mu * mu;
    float sc = gamma[d] * rsqrtf(var + BN_EPS);
    stats[d] = sc;
    stats[D + d] = beta[d] - mu * sc;
  }
}

__global__ void bn_apply_kernel(const float* __restrict__ h2,
                                const float* __restrict__ stats,
                                float* __restrict__ dst, int relu) {
  int i = blockIdx.x * blockDim.x + threadIdx.x;
  if (i >= NN * D) return;
  int d = i & (D - 1);
  float v = stats[d] * h2[i] + stats[D + d];
  if (relu) v = fmaxf(v, 0.f);
  dst[i] = v;
}

extern "C" void kernel_launch(void* const* d_in, const int* in_sizes, int n_in,
                              void* d_out, int out_size, void* d_ws, size_t ws_size,
                              hipStream_t stream) {
  const int*   x     = (const int*)d_in[0];
  const int*   ei    = (const int*)d_in[1];
  const int*   ea    = (const int*)d_in[2];
  const float* xemb1 = (const float*)d_in[4];
  const float* xemb2 = (const float*)d_in[5];
  const float* e1    = (const float*)d_in[6];
  const float* e2    = (const float*)d_in[7];
  const float* W1    = (const float*)d_in[8];
  const float* b1    = (const float*)d_in[9];
  const float* W2    = (const float*)d_in[10];
  const float* b2    = (const float*)d_in[11];
  const float* gamma = (const float*)d_in[12];
  const float* beta  = (const float*)d_in[13];
  float* out = (float*)d_out;

  float* ws   = (float*)d_ws;
  float* h    = ws;                          // [N, D]
  float* agg  = ws + (size_t)NN * D;         // [N, D]  (reused as h2)
  float* h2a  = ws + 2 * (size_t)NN * D;     // [N, 2D]
  float* h2   = agg;
  float* stats = ws + 4 * (size_t)NN * D;    // [2, D]

  const int nd = NN * D;
  const int eb = (nd + 255) / 256;

  embed_kernel<<<eb, 256, 0, stream>>>(x, xemb1, xemb2, h);

  for (int l = 0; l < 2; ++l) {
    const float* e1l = e1 + (size_t)l * 7 * D;
    const float* e2l = e2 + (size_t)l * 3 * D;

    init_agg_kernel<<<eb, 256, 0, stream>>>(h, e1l, e2l, agg);
    scatter_kernel<<<(NE * 32 + 255) / 256, 256, 0, stream>>>(ei, ea, h, e1l, e2l, agg);

    // GEMM1: [N,128] @ [128,256] + b1, relu  -> h2a
    int waves1 = (NN / 16) * (D2 / 64);
    wmma_gemm_bias<4><<<(waves1 * 32 + 255) / 256, 256, 0, stream>>>(
        agg, W1 + (size_t)l * D * D2, b1 + (size_t)l * D2, h2a, D, D2, waves1, 1);

    // GEMM2: [N,256] @ [256,128] + b2 -> h2 (= agg buffer)
    int waves2 = (NN / 16) * (D / 64);
    wmma_gemm_bias<4><<<(waves2 * 32 + 255) / 256, 256, 0, stream>>>(
        h2a, W2 + (size_t)l * D2 * D, b2 + (size_t)l * D, h2, D2, D, waves2, 0);

    bn_stats_kernel<<<D, 256, 0, stream>>>(h2, gamma + (size_t)l * D, beta + (size_t)l * D, stats);

    float* dst = (l == 0) ? h : out;
    bn_apply_kernel<<<eb, 256, 0, stream>>>(h2, stats, dst, (l == 0) ? 1 : 0);
  }
}